// GcnNet_64991445123420
// MI455X (gfx1250) — compile-verified
//
#include <hip/hip_runtime.h>
#include <hip/hip_bf16.h>

typedef __attribute__((ext_vector_type(2))) float v2f;
typedef __attribute__((ext_vector_type(8))) float v8f;
typedef __attribute__((ext_vector_type(4))) unsigned int u32x4;
typedef __attribute__((ext_vector_type(4))) int i32x4;
typedef __attribute__((ext_vector_type(8))) int i32x8;

#define NN   750000
#define EE   6000000
#define GG   10000
#define HH   64
#define BN_EPS 1e-5f

// ---------------- utility ----------------
__global__ void fill_f32(float* __restrict__ p, float v, int n) {
  int i = blockIdx.x * blockDim.x + threadIdx.x;
  if (i < n) p[i] = v;
}

// deg[dst] += 1  (deg pre-filled with 1.0 for the self loop)
__global__ void deg_accum(const int* __restrict__ dst, float* __restrict__ deg, int e) {
  int i = blockIdx.x * blockDim.x + threadIdx.x;
  if (i < e) atomicAdd(&deg[dst[i]], 1.0f);
}

__global__ void deg_to_dinv(float* __restrict__ d, int n) {
  int i = blockIdx.x * blockDim.x + threadIdx.x;
  if (i < n) { float x = d[i]; d[i] = (x > 0.0f) ? rsqrtf(x) : 0.0f; }
}

// ---------------- layer 1 linear: [N,2] @ [2,64], fused accumulator init ----------------
// tmp[i,c]  = x[i,:] @ W1[:,c]
// hout[i,c] = b[c] + tmp[i,c] * dinv[i]^2   (bias + self-loop term)
__global__ void lin1(const float* __restrict__ x, const float* __restrict__ W1,
                     const float* __restrict__ b, const float* __restrict__ dinv,
                     float* __restrict__ tmp, float* __restrict__ hout, int n) {
  int i = blockIdx.x * blockDim.x + threadIdx.x;   // over n*64
  if (i < n * 64) {
    int node = i >> 6, c = i & 63;
    float t = x[node * 2] * W1[c] + x[node * 2 + 1] * W1[64 + c];
    float di = dinv[node];
    tmp[i] = t;
    hout[i] = b[c] + t * di * di;
  }
}

// ---------------- layers 2/3 linear: relu(hin) [N,64] @ W [64,64] via WMMA f32 --------
// Block = 256 threads = 8 waves; each wave owns a 16-row tile -> 128 rows/block.
// W is DMA'd into LDS by the Tensor Data Mover, then transposed in LDS so every
// B fragment (two consecutive K values of one column) is an 8-byte ds_load_b64.
// A tile (16x4 f32): lane m=lane&15 holds row m; VGPR0/1 = K {0,1} (lanes 0-15)
// or K {2,3} (lanes 16-31).  B tile (4x16): same K split, N = lane&15.
// D (16x16 f32): VGPR r, lanes 0-15 -> M=r, lanes 16-31 -> M=r+8.
// Fused: relu on A staging, and hout[i,c] = bias[c] + D[i,c]*dinv[i]^2 on store.
__global__ __launch_bounds__(256) void lin64_wmma(const float* __restrict__ hin,
                                                  const float* __restrict__ W,
                                                  const float* __restrict__ bias,
                                                  const float* __restrict__ dinv,
                                                  float* __restrict__ tmp,
                                                  float* __restrict__ hout, int n) {
  __shared__ float sWraw[64 * 64];  // 16 KB, row-major W (K x N), TDM destination
  __shared__ float sWt[64 * 64];    // 16 KB, transposed: sWt[n*64 + k]
  __shared__ float sA[128 * 64];    // 32 KB
  int tid = threadIdx.x;

  // ---- stage W into LDS via the Tensor Data Mover (wave 0 only; TDM ignores EXEC,
  //      the branch is wave-uniform so full waves skip it) ----
#if __has_builtin(__builtin_amdgcn_tensor_load_to_lds)
  if (tid < 32) {
    unsigned lds_off = (unsigned)(unsigned long long)(uintptr_t)sWraw; // low 32b of flat = LDS offset
    unsigned long long ga = (unsigned long long)(uintptr_t)W;
    u32x4 g0;
    g0[0] = 1u;                                                // count=1, user D#, no gather
    g0[1] = lds_off;                                           // lds_addr
    g0[2] = (unsigned)(ga & 0xFFFFFFFFu);                      // global_addr[31:0]
    g0[3] = (unsigned)((ga >> 32) & 0x01FFFFFFu) | 0x80000000u;// global_addr[56:32] | type=2
    i32x8 g1;
    g1[0] = (2 << 16);   // workgroup_mask=0, data_size=4B, no barrier/iterate/pad
    g1[1] = (64 << 16);  // tensor_dim0 = 64
    g1[2] = (64 << 16);  // tensor_dim1[15:0] = 64
    g1[3] = (64 << 16);  // tile_dim0 = 64
    g1[4] = 64;          // tile_dim1 = 64, tile_dim2 = 0
    g1[5] = 64;          // tensor_dim0_stride = 64
    g1[6] = 0;
    g1[7] = 0;
    i32x4 z4 = {0, 0, 0, 0};
#if __clang_major__ >= 23
    i32x8 z8 = {0, 0, 0, 0, 0, 0, 0, 0};
    __builtin_amdgcn_tensor_load_to_lds(g0, g1, z4, z4, z8, 0);
#else
    __builtin_amdgcn_tensor_load_to_lds(g0, g1, z4, z4, 0);
#endif
    __builtin_amdgcn_s_wait_tensorcnt(0);
  }
#else
  for (int idx = tid; idx < 64 * 64; idx += 256) sWraw[idx] = W[idx];
#endif

  // ---- stage A tile (relu fused: hin holds pre-activation of previous layer) ----
  int rbase = blockIdx.x * 128;
  for (int idx = tid; idx < 128 * 64; idx += 256) {
    int row = rbase + (idx >> 6);
    sA[idx] = (row < n) ? fmaxf(hin[row * 64 + (idx & 63)], 0.0f) : 0.0f;
  }
  __syncthreads();

  // ---- transpose W inside LDS so B fetches are contiguous b64 loads ----
  for (int idx = tid; idx < 64 * 64; idx += 256) {
    int kk = idx >> 6, cc = idx & 63;
    sWt[cc * 64 + kk] = sWraw[idx];
  }
  __syncthreads();

  int wave = tid >> 5;
  int lane = tid & 31;
  int rowbase = rbase + wave * 16;          // N % 16 == 0 -> tiles fully in/out
  if (rowbase >= n) return;                 // wave-uniform: EXEC stays all-ones

  int m = lane & 15;
  int half = lane >> 4;
  v8f acc0 = {}, acc1 = {}, acc2 = {}, acc3 = {};
  const float* aRow  = &sA[(wave * 16 + m) * 64];
  const float* bCol0 = &sWt[m * 64];        // columns m, 16+m, 32+m, 48+m

  #pragma unroll
  for (int kt = 0; kt < 16; ++kt) {
    int k = kt * 4 + 2 * half;              // k is even -> 8B-aligned LDS loads
    v2f a  = *(const v2f*)&aRow[k];
    v2f b0 = *(const v2f*)&bCol0[k];
    v2f b1 = *(const v2f*)&bCol0[16 * 64 + k];
    v2f b2 = *(const v2f*)&bCol0[32 * 64 + k];
    v2f b3 = *(const v2f*)&bCol0[48 * 64 + k];
    acc0 = __builtin_amdgcn_wmma_f32_16x16x4_f32(false, a, false, b0, (short)0, acc0, false, false);
    acc1 = __builtin_amdgcn_wmma_f32_16x16x4_f32(false, a, false, b1, (short)0, acc1, false, false);
    acc2 = __builtin_amdgcn_wmma_f32_16x16x4_f32(false, a, false, b2, (short)0, acc2, false, false);
    acc3 = __builtin_amdgcn_wmma_f32_16x16x4_f32(false, a, false, b3, (short)0, acc3, false, false);
  }

  float bias0 = bias[m], bias1 = bias[16 + m], bias2 = bias[32 + m], bias3 = bias[48 + m];
  #pragma unroll
  for (int r = 0; r < 8; ++r) {
    int row = rowbase + r + 8 * half;
    float di = dinv[row];
    float d2 = di * di;
    float* o  = &tmp[row * 64 + m];
    float* ho = &hout[row * 64 + m];
    o[0]  = acc0[r];  ho[0]  = bias0 + acc0[r] * d2;
    o[16] = acc1[r];  ho[16] = bias1 + acc1[r] * d2;
    o[32] = acc2[r];  ho[32] = bias2 + acc2[r] * d2;
    o[48] = acc3[r];  ho[48] = bias3 + acc3[r] * d2;
  }
}

// per (edge, 4-channel group): hout[dst] += tmp[src] * dinv[src]*dinv[dst]
__global__ void scatter_edges(const int* __restrict__ src, const int* __restrict__ dst,
                              const float* __restrict__ dinv, const float* __restrict__ tmp,
                              float* __restrict__ hout, int e) {
  int t = blockIdx.x * blockDim.x + threadIdx.x;
  int eidx = t >> 4;
  if (eidx >= e) return;
  int c4 = (t & 15) * 4;
  int s = src[eidx], d = dst[eidx];
  float norm = dinv[s] * dinv[d];
  const float4 v = *(const float4*)&tmp[s * 64 + c4];
  float* o = &hout[d * 64 + c4];
  atomicAdd(o + 0, v.x * norm);
  atomicAdd(o + 1, v.y * norm);
  atomicAdd(o + 2, v.z * norm);
  atomicAdd(o + 3, v.w * norm);
}

// ---------------- pooling (relu of layer-3 pre-activation fused in) ----------------
__global__ void pool_accum(const float* __restrict__ h, const int* __restrict__ batch,
                           float* __restrict__ sums, float* __restrict__ cnts, int n) {
  int t = blockIdx.x * blockDim.x + threadIdx.x;
  int node = t >> 4;
  if (node >= n) return;
  int c4 = (t & 15) * 4;
  int g = batch[node];
  const float4 v = *(const float4*)&h[node * 64 + c4];
  float* o = &sums[g * 64 + c4];
  atomicAdd(o + 0, fmaxf(v.x, 0.0f));
  atomicAdd(o + 1, fmaxf(v.y, 0.0f));
  atomicAdd(o + 2, fmaxf(v.z, 0.0f));
  atomicAdd(o + 3, fmaxf(v.w, 0.0f));
  if ((t & 15) == 0) atomicAdd(&cnts[g], 1.0f);
}

__global__ void pooled_div(float* __restrict__ sums, const float* __restrict__ cnts, int n) {
  int i = blockIdx.x * blockDim.x + threadIdx.x;   // over G*64
  if (i < n) sums[i] /= fmaxf(cnts[i >> 6], 1.0f);
}

// one block per channel; biased variance -> fold gamma/beta into scale/shift
__global__ __launch_bounds__(256) void bn_stats(const float* __restrict__ pooled,
                                                const float* __restrict__ gamma,
                                                const float* __restrict__ beta,
                                                float* __restrict__ scale,
                                                float* __restrict__ shift) {
  int c = blockIdx.x;
  __shared__ float ssum[256], ssq[256];
  float s = 0.0f, q = 0.0f;
  for (int g = threadIdx.x; g < GG; g += 256) {
    float v = pooled[g * 64 + c];
    s += v; q += v * v;
  }
  ssum[threadIdx.x] = s; ssq[threadIdx.x] = q;
  __syncthreads();
  for (int off = 128; off > 0; off >>= 1) {
    if (threadIdx.x < off) {
      ssum[threadIdx.x] += ssum[threadIdx.x + off];
      ssq[threadIdx.x]  += ssq[threadIdx.x + off];
    }
    __syncthreads();
  }
  if (threadIdx.x == 0) {
    float mean = ssum[0] / (float)GG;
    float var  = ssq[0] / (float)GG - mean * mean;
    float sc = gamma[c] * rsqrtf(var + BN_EPS);
    scale[c] = sc;
    shift[c] = beta[c] - mean * sc;
  }
}

// ---------------- MLP head + log_softmax (one thread per graph) ----------------
__global__ __launch_bounds__(256) void fc_head(const float* __restrict__ pooled,
                                               const float* __restrict__ scale,
                                               const float* __restrict__ shift,
                                               const float* __restrict__ fcW1,
                                               const float* __restrict__ fcb1,
                                               const float* __restrict__ fcW2,
                                               const float* __restrict__ fcb2,
                                               float* __restrict__ out, int g_total) {
  __shared__ float sW1[64 * 32];
  __shared__ float sW2[32 * 10];
  __shared__ float sb1[32], sb2[10], sscale[64], sshift[64];
  int tid = threadIdx.x;
  for (int i = tid; i < 64 * 32; i += 256) sW1[i] = fcW1[i];
  for (int i = tid; i < 320; i += 256) sW2[i] = fcW2[i];
  if (tid < 32) sb1[tid] = fcb1[tid];
  if (tid < 10) sb2[tid] = fcb2[tid];
  if (tid < 64) { sscale[tid] = scale[tid]; sshift[tid] = shift[tid]; }
  __syncthreads();
  int g = blockIdx.x * 256 + tid;
  if (g >= g_total) return;

  float hbn[64];
  #pragma unroll
  for (int c = 0; c < 64; ++c) hbn[c] = pooled[g * 64 + c] * sscale[c] + sshift[c];

  float h1[32];
  #pragma unroll
  for (int j = 0; j < 32; ++j) {
    float s = sb1[j];
    for (int c = 0; c < 64; ++c) s += hbn[c] * sW1[c * 32 + j];
    h1[j] = fmaxf(s, 0.0f);
  }

  float logit[10];
  float mx = -3.0e38f;
  #pragma unroll
  for (int k = 0; k < 10; ++k) {
    float s = sb2[k];
    for (int j = 0; j < 32; ++j) s += h1[j] * sW2[j * 10 + k];
    logit[k] = s;
    mx = fmaxf(mx, s);
  }
  float sum = 0.0f;
  #pragma unroll
  for (int k = 0; k < 10; ++k) sum += expf(logit[k] - mx);
  float lse = mx + logf(sum);
  #pragma unroll
  for (int k = 0; k < 10; ++k) out[g * 10 + k] = logit[k] - lse;
}

// ---------------- host orchestration ----------------
extern "C" void kernel_launch(void* const* d_in, const int* in_sizes, int n_in,
                              void* d_out, int out_size, void* d_ws, size_t ws_size,
                              hipStream_t stream) {
  const float* x     = (const float*)d_in[0];
  const int*   ei    = (const int*)  d_in[1];
  const int*   batch = (const int*)  d_in[2];
  const float* W1    = (const float*)d_in[3];
  const float* b1    = (const float*)d_in[4];
  const float* W2    = (const float*)d_in[5];
  const float* b2    = (const float*)d_in[6];
  const float* W3    = (const float*)d_in[7];
  const float* b3    = (const float*)d_in[8];
  const float* gamma = (const float*)d_in[9];
  const float* beta  = (const float*)d_in[10];
  const float* fcW1  = (const float*)d_in[11];
  const float* fcb1  = (const float*)d_in[12];
  const float* fcW2  = (const float*)d_in[13];
  const float* fcb2  = (const float*)d_in[14];
  const int* src = ei;
  const int* dst = ei + EE;
  float* out = (float*)d_out;

  char* ws = (char*)d_ws;
  size_t off = 0;
  auto carve = [&](size_t bytes) -> float* {
    float* p = (float*)(ws + off);
    off = (off + bytes + 255) & ~(size_t)255;
    return p;
  };
  float* dinv  = carve((size_t)NN * 4);          // degree -> dinv, in place
  float* buf0  = carve((size_t)NN * 64 * 4);     // tmp = h @ W
  float* buf1  = carve((size_t)NN * 64 * 4);     // scatter accumulator / layer output
  float* sums  = carve((size_t)GG * 64 * 4);     // pooled sums -> pooled
  float* cnts  = carve((size_t)GG * 4);
  float* scale = carve(64 * 4);
  float* shift = carve(64 * 4);

  const int B = 256;
  const int nh = NN * 64;
  const int gbN   = (NN + B - 1) / B;
  const int gbE   = (EE + B - 1) / B;
  const int gbNH  = (nh + B - 1) / B;
  const int gbE16 = (EE * 16 + B - 1) / B;
  const int gbN16 = (NN * 16 + B - 1) / B;
  const int gbWM  = (NN + 127) / 128;

  // degrees (self loop contributes 1.0 to every node)
  fill_f32<<<gbN, B, 0, stream>>>(dinv, 1.0f, NN);
  deg_accum<<<gbE, B, 0, stream>>>(dst, dinv, EE);
  deg_to_dinv<<<gbN, B, 0, stream>>>(dinv, NN);

  // ---- layer 1: tmp=x@W1 (+fused accumulator init), scatter ----
  lin1<<<gbNH, B, 0, stream>>>(x, W1, b1, dinv, buf0, buf1, NN);
  scatter_edges<<<gbE16, B, 0, stream>>>(src, dst, dinv, buf0, buf1, EE);

  // ---- layer 2: tmp=relu(h1)@W2 via WMMA (+fused init), scatter ----
  lin64_wmma<<<gbWM, B, 0, stream>>>(buf1, W2, b2, dinv, buf0, buf1, NN);
  scatter_edges<<<gbE16, B, 0, stream>>>(src, dst, dinv, buf0, buf1, EE);

  // ---- layer 3 ----
  lin64_wmma<<<gbWM, B, 0, stream>>>(buf1, W3, b3, dinv, buf0, buf1, NN);
  scatter_edges<<<gbE16, B, 0, stream>>>(src, dst, dinv, buf0, buf1, EE);

  // ---- pooling (relu fused) + BN + head ----
  fill_f32<<<(GG * 64 + B - 1) / B, B, 0, stream>>>(sums, 0.0f, GG * 64);
  fill_f32<<<(GG + B - 1) / B, B, 0, stream>>>(cnts, 0.0f, GG);
  pool_accum<<<gbN16, B, 0, stream>>>(buf1, batch, sums, cnts, NN);
  pooled_div<<<(GG * 64 + B - 1) / B, B, 0, stream>>>(sums, cnts, GG * 64);
  bn_stats<<<64, 256, 0, stream>>>(sums, gamma, beta, scale, shift);
  fc_head<<<(GG + 255) / 256, 256, 0, stream>>>(sums, scale, shift,
                                                fcW1, fcb1, fcW2, fcb2, out, GG);
}